// GCNNet_14680198218267
// MI455X (gfx1250) — compile-verified
//
#include <hip/hip_runtime.h>
#include <hip/hip_bf16.h>

typedef float v2f __attribute__((ext_vector_type(2)));
typedef float v8f __attribute__((ext_vector_type(8)));

#define D 128
#define G 64

// ---------------- GEMM: H = X(N x 128) * W(128 x 128) via V_WMMA_F32_16X16X4_F32
// W is staged in LDS with row-pairs interleaved: Wl[k2*128 + n] = {W[2k2][n], W[2k2+1][n]}
// so each lane's B fragment is ONE aligned ds_load_b64 (no repacking movs).
__global__ void __launch_bounds__(256) gemm_wmma_kernel(const float* __restrict__ X,
                                                        const float* __restrict__ W,
                                                        float* __restrict__ H,
                                                        int nTiles) {
    __shared__ v2f Wl[64 * D];  // 64 KB
    for (int idx = threadIdx.x; idx < 64 * D; idx += 256) {
        int k2 = idx >> 7, n = idx & (D - 1);
        v2f w;
        w.x = W[(2 * k2) * D + n];
        w.y = W[(2 * k2 + 1) * D + n];
        Wl[idx] = w;
    }
    __syncthreads();

    const int wave = threadIdx.x >> 5;
    const int lane = threadIdx.x & 31;
    const int tile = blockIdx.x * 8 + wave;
    if (tile >= nTiles) return;           // wave-uniform: EXEC all-ones inside
    const int r0    = tile * 16;
    const int laneR = lane & 15;
    const int hi    = lane >> 4;          // 0: K pair {0,1}, 1: K pair {2,3}

    v8f acc[8];
#pragma unroll
    for (int t = 0; t < 8; ++t) acc[t] = (v8f){0.f,0.f,0.f,0.f,0.f,0.f,0.f,0.f};

    // A fragment: row = r0+laneR, columns (4*kk + 2*hi, +1) -> single global b64 load
    const float* xrow = X + (size_t)(r0 + laneR) * D + 2 * hi;
    // B fragment base: pair index (2*kk + hi)*128 + nt*16 + laneR
    const v2f* wbase = Wl + hi * D + laneR;

#pragma unroll 4
    for (int kk = 0; kk < 32; ++kk) {
        v2f a = *(const v2f*)(xrow + kk * 4);
        const v2f* wp = wbase + kk * (2 * D);
#pragma unroll
        for (int nt = 0; nt < 8; ++nt) {
            v2f b = wp[nt * 16];
            acc[nt] = __builtin_amdgcn_wmma_f32_16x16x4_f32(
                false, a, false, b, (short)0, acc[nt], false, false);
        }
    }

    // C/D layout: VGPR j -> M = j + 8*hi, N = laneR
    const int rowBase = r0 + hi * 8;
#pragma unroll
    for (int nt = 0; nt < 8; ++nt) {
#pragma unroll
        for (int j = 0; j < 8; ++j) {
            H[(size_t)(rowBase + j) * D + nt * 16 + laneR] = acc[nt][j];
        }
    }
}

// ---------------- small utility kernels
__global__ void zero_f32_kernel(float* __restrict__ p, int n) {
    int i = blockIdx.x * blockDim.x + threadIdx.x;
    if (i < n) p[i] = 0.0f;
}

__global__ void bounds_init_kernel(int* __restrict__ gstart, int* __restrict__ gend, int n) {
    int g = threadIdx.x;
    if (g < G) { gstart[g] = n; gend[g] = -1; }
}

__global__ void bounds_kernel(const int* __restrict__ batch, int* __restrict__ gstart,
                              int* __restrict__ gend, int n) {
    int i = blockIdx.x * blockDim.x + threadIdx.x;
    if (i < n) {
        int g = batch[i];
        atomicMin(&gstart[g], i);
        atomicMax(&gend[g], i);
    }
}

__global__ void counts_kernel(const int* __restrict__ gstart, const int* __restrict__ gend,
                              float* __restrict__ counts) {
    int g = threadIdx.x;
    if (g < G) {
        int c = gend[g] - gstart[g] + 1;
        counts[g] = (float)(c > 1 ? c : 1);
    }
}

__global__ void deg_edges_kernel(const int* __restrict__ dst, float* __restrict__ deg, int E) {
    int e = blockIdx.x * blockDim.x + threadIdx.x;
    if (e < E) atomicAdd(&deg[dst[e]], 1.0f);
}

__global__ void dinv_kernel(const float* __restrict__ deg, float* __restrict__ dinv, int n) {
    int i = blockIdx.x * blockDim.x + threadIdx.x;
    if (i < n) dinv[i] = rsqrtf(deg[i] + 1.0f);   // +1 = self loop; always > 0
}

// agg = bias + h * dinv^2   (self-loop contribution; full write, no memset needed)
__global__ void init_agg_kernel(const float* __restrict__ h, const float* __restrict__ dinv,
                                const float* __restrict__ bias, float* __restrict__ agg, int nd) {
    int idx = blockIdx.x * blockDim.x + threadIdx.x;
    if (idx >= nd) return;
    int i = idx >> 7, d = idx & (D - 1);
    float di = dinv[i];
    agg[idx] = bias[d] + h[idx] * di * di;
}

// one wave per edge, float4 per lane
__global__ void __launch_bounds__(256) edge_scatter_kernel(const float* __restrict__ H,
                                                           const int* __restrict__ src,
                                                           const int* __restrict__ dst,
                                                           const float* __restrict__ dinv,
                                                           float* __restrict__ agg, int E) {
    int lane = threadIdx.x & 31;
    int e = (blockIdx.x * 256 + threadIdx.x) >> 5;
    if (e >= E) return;
    int s = src[e], t = dst[e];
    float w = dinv[s] * dinv[t];
    const float4 hv = *(const float4*)(H + (size_t)s * D + lane * 4);
    float* ap = agg + (size_t)t * D + lane * 4;
    atomicAdd(ap + 0, hv.x * w);
    atomicAdd(ap + 1, hv.y * w);
    atomicAdd(ap + 2, hv.z * w);
    atomicAdd(ap + 3, hv.w * w);
}

// block-per-graph segmented mean (sorted batch -> contiguous ranges), no atomics
__global__ void __launch_bounds__(256) graph_mean_kernel(const float* __restrict__ agg,
                                                         const int* __restrict__ gstart,
                                                         const int* __restrict__ gend,
                                                         const float* __restrict__ counts,
                                                         float* __restrict__ mean) {
    __shared__ float part[256];
    int g = blockIdx.x;
    int d = threadIdx.x & (D - 1);
    int rg = threadIdx.x >> 7;   // 0/1
    int i0 = gstart[g], i1 = gend[g];
    float sum = 0.0f;
    for (int i = i0 + rg; i <= i1; i += 2) sum += agg[(size_t)i * D + d];
    part[threadIdx.x] = sum;
    __syncthreads();
    if (threadIdx.x < D) mean[g * D + d] = (part[d] + part[d + D]) / counts[g];
}

// centered = agg - alpha*mean (stored back into agg); rstd = rsqrt(var + eps)
__global__ void __launch_bounds__(256) center_var_kernel(float* __restrict__ agg,
                                                         const float* __restrict__ mean,
                                                         const float* __restrict__ alpha,
                                                         const int* __restrict__ gstart,
                                                         const int* __restrict__ gend,
                                                         const float* __restrict__ counts,
                                                         float* __restrict__ rstd) {
    __shared__ float part[256];
    int g = blockIdx.x;
    int d = threadIdx.x & (D - 1);
    int rg = threadIdx.x >> 7;
    float am = alpha[d] * mean[g * D + d];
    int i0 = gstart[g], i1 = gend[g];
    float vs = 0.0f;
    for (int i = i0 + rg; i <= i1; i += 2) {
        size_t o = (size_t)i * D + d;
        float c = agg[o] - am;
        agg[o] = c;
        vs += c * c;
    }
    part[threadIdx.x] = vs;
    __syncthreads();
    if (threadIdx.x < D) {
        float var = (part[d] + part[d + D]) / counts[g];
        rstd[g * D + d] = rsqrtf(var + 1e-5f);
    }
}

// x += relu(gamma * centered * rstd + beta)
__global__ void finalize_kernel(const float* __restrict__ agg, const float* __restrict__ rstd,
                                const float* __restrict__ gamma, const float* __restrict__ beta,
                                const int* __restrict__ batch, float* __restrict__ x, int nd) {
    int idx = blockIdx.x * blockDim.x + threadIdx.x;
    if (idx >= nd) return;
    int i = idx >> 7, d = idx & (D - 1);
    int g = batch[i];
    float v = gamma[d] * agg[idx] * rstd[g * D + d] + beta[d];
    x[idx] += fmaxf(v, 0.0f);
}

// mean pool per graph -> out[G, D]
__global__ void __launch_bounds__(256) pool_kernel(const float* __restrict__ x,
                                                   const int* __restrict__ gstart,
                                                   const int* __restrict__ gend,
                                                   const float* __restrict__ counts,
                                                   float* __restrict__ out) {
    __shared__ float part[256];
    int g = blockIdx.x;
    int d = threadIdx.x & (D - 1);
    int rg = threadIdx.x >> 7;
    int i0 = gstart[g], i1 = gend[g];
    float sum = 0.0f;
    for (int i = i0 + rg; i <= i1; i += 2) sum += x[(size_t)i * D + d];
    part[threadIdx.x] = sum;
    __syncthreads();
    if (threadIdx.x < D) out[g * D + d] = (part[d] + part[d + D]) / counts[g];
}

extern "C" void kernel_launch(void* const* d_in, const int* in_sizes, int n_in,
                              void* d_out, int out_size, void* d_ws, size_t ws_size,
                              hipStream_t stream) {
    const float* x_in   = (const float*)d_in[0];
    const int*   ei     = (const int*)d_in[1];
    const int*   batch  = (const int*)d_in[2];
    const float* Ws     = (const float*)d_in[3];
    const float* bs     = (const float*)d_in[4];
    const float* gammas = (const float*)d_in[5];
    const float* betas  = (const float*)d_in[6];
    const float* alphas = (const float*)d_in[7];

    const int N = in_sizes[2];         // 50000
    const int E = in_sizes[1] / 2;     // undirected coalesced edges
    const int ND = N * D;
    const int* src = ei;
    const int* dst = ei + E;

    // workspace carve-up (all 16B aligned)
    char* p = (char*)d_ws;
    float* xbuf   = (float*)p; p += (size_t)ND * 4;
    float* h      = (float*)p; p += (size_t)ND * 4;
    float* agg    = (float*)p; p += (size_t)ND * 4;
    float* deg    = (float*)p; p += (size_t)N * 4;
    float* dinv   = (float*)p; p += (size_t)N * 4;
    float* mean   = (float*)p; p += (size_t)G * D * 4;
    float* rstd   = (float*)p; p += (size_t)G * D * 4;
    float* counts = (float*)p; p += 256;
    int*   gstart = (int*)p;   p += 256;
    int*   gend   = (int*)p;   p += 256;

    const int B = 256;
    const int gN  = (N + B - 1) / B;
    const int gND = (ND + B - 1) / B;
    const int gE  = (E + B - 1) / B;

    hipMemcpyAsync(xbuf, x_in, (size_t)ND * 4, hipMemcpyDeviceToDevice, stream);

    // graph ranges + counts (batch is sorted -> contiguous segments)
    bounds_init_kernel<<<1, G, 0, stream>>>(gstart, gend, N);
    bounds_kernel<<<gN, B, 0, stream>>>(batch, gstart, gend, N);
    counts_kernel<<<1, G, 0, stream>>>(gstart, gend, counts);

    // degree (incl. self loop) -> dinv
    zero_f32_kernel<<<gN, B, 0, stream>>>(deg, N);
    deg_edges_kernel<<<gE, B, 0, stream>>>(dst, deg, E);
    dinv_kernel<<<gN, B, 0, stream>>>(deg, dinv, N);

    const int nTiles = N / 16;                       // 50000 % 16 == 0
    const int gGemm  = (nTiles + 7) / 8;
    const int gScat  = (E * 32 + B - 1) / B;

    for (int l = 0; l < 3; ++l) {
        gemm_wmma_kernel<<<gGemm, B, 0, stream>>>(xbuf, Ws + (size_t)l * D * D, h, nTiles);
        init_agg_kernel<<<gND, B, 0, stream>>>(h, dinv, bs + l * D, agg, ND);
        edge_scatter_kernel<<<gScat, B, 0, stream>>>(h, src, dst, dinv, agg, E);
        graph_mean_kernel<<<G, B, 0, stream>>>(agg, gstart, gend, counts, mean);
        center_var_kernel<<<G, B, 0, stream>>>(agg, mean, alphas + l * D, gstart, gend,
                                               counts, rstd);
        finalize_kernel<<<gND, B, 0, stream>>>(agg, rstd, gammas + l * D, betas + l * D,
                                               batch, xbuf, ND);
    }

    pool_kernel<<<G, B, 0, stream>>>(xbuf, gstart, gend, counts, (float*)d_out);
}